// CoordinateDescent_48490180772168
// MI455X (gfx1250) — compile-verified
//
#include <hip/hip_runtime.h>

// CDNA5 / gfx1250. Wave32. Uses V_WMMA_F32_16X16X4_F32 for the two
// tall-skinny fp32 GEMMs (P = X@V and P2 = X^T@U_new); the coordinate-descent
// recurrence collapses to a per-row 16-step scan using the 16x16 Gram matrix.

typedef __attribute__((ext_vector_type(2))) float v2f;
typedef __attribute__((ext_vector_type(8))) float v8f;

#define EPSF 1e-8f

constexpr int Bc = 2;
constexpr int Mc = 2048;
constexpr int Nc = 2048;
constexpr int Rr = 16;
constexpr int KC = 128;                   // k-chunk of B-operand staged in LDS (8 KB)
constexpr int WAVES = 8;                  // 256 threads / 32
constexpr int ROWS_PER_WG = WAVES * 16;   // 128 output rows per workgroup

// ---------------------------------------------------------------------------
// G[b] = W^T W (16x16), W: [L x 16] row-major per batch.
// One workgroup per batch; thread t owns G[i=t/16][j=t%16].
__global__ void __launch_bounds__(256)
gram16(const float* __restrict__ W, float* __restrict__ G, int L) {
  const int b = blockIdx.x;
  const int t = threadIdx.x;
  const int i = t >> 4, j = t & 15;
  __shared__ float s[256];
  const float* Wb = W + (size_t)b * L * 16;
  float acc = 0.f;
  for (int n0 = 0; n0 < L; n0 += 16) {
    s[t] = Wb[n0 * 16 + t];               // 16 rows x 16 cols, coalesced
    __syncthreads();
#pragma unroll
    for (int k = 0; k < 16; ++k)
      acc = __builtin_fmaf(s[k * 16 + i], s[k * 16 + j], acc);
    __syncthreads();
  }
  G[b * 256 + t] = acc;
}

// ---------------------------------------------------------------------------
// P[b] = op(X[b]) @ W[b],  op = identity (TRANS=0, Mo=M, Kc=N)
//                          op = transpose (TRANS=1, Mo=N, Kc=M)
// X: [B, Mc, Nc] row-major.  W: [B, Kc, 16].  P: [B, Mo, 16].
// One wave per 16-row output tile; W chunk shared via LDS.
template <bool TRANS>
__global__ void __launch_bounds__(256)
gemm_x_w(const float* __restrict__ X, const float* __restrict__ W,
         float* __restrict__ P, int Mo, int Kc) {
  const int b    = blockIdx.y;
  const int tid  = threadIdx.x;
  const int wave = tid >> 5;
  const int lane = tid & 31;
  const int half = lane >> 4;             // 0: lanes 0-15, 1: lanes 16-31
  const int mi   = lane & 15;
  const int m0   = blockIdx.x * ROWS_PER_WG + wave * 16;

  const float* Xb = X + (size_t)b * Mc * Nc;
  const float* Wb = W + (size_t)b * Kc * 16;
  float*       Pb = P + (size_t)b * Mo * 16;

  __shared__ float sw[KC * 16];

  v8f acc = {};
  for (int k0 = 0; k0 < Kc; k0 += KC) {
    // Stage W chunk: KC*16 contiguous floats, fully coalesced.
#pragma unroll
    for (int i = 0; i < (KC * 16) / 256; ++i)
      sw[tid + i * 256] = Wb[k0 * 16 + tid + i * 256];
    __syncthreads();

    if (k0 + KC < Kc) {                   // gfx1250 global_prefetch_b8 on next X chunk
      const float* nxt = TRANS ? (Xb + (size_t)(k0 + KC) * Nc + m0 + mi)
                               : (Xb + (size_t)(m0 + mi) * Nc + (k0 + KC));
      __builtin_prefetch(nxt, 0, 1);
    }

#pragma unroll 4
    for (int kk = 0; kk < KC; kk += 4) {
      const int k = kk + 2 * half;        // this half-wave's K pair within the 4-step
      // A fragment: lane holds A[m0+mi, k0+k] and A[m0+mi, k0+k+1]
      v2f a;
      if (!TRANS) {
        const float* px = Xb + (size_t)(m0 + mi) * Nc + (k0 + k);
        a.x = px[0];
        a.y = px[1];                      // adjacent columns (8B-aligned pair)
      } else {
        const float* px = Xb + (size_t)(k0 + k) * Nc + (m0 + mi);
        a.x = px[0];
        a.y = px[Nc];                     // adjacent rows; coalesced across lanes
      }
      // B fragment: lane holds W[k0+k, mi] and W[k0+k+1, mi] from LDS
      v2f bf;
      bf.x = sw[(k)     * 16 + mi];
      bf.y = sw[(k + 1) * 16 + mi];
      acc = __builtin_amdgcn_wmma_f32_16x16x4_f32(false, a, false, bf,
                                                  (short)0, acc, false, false);
    }
    __syncthreads();
  }

  // D layout: VGPR j -> row (j + 8*half), col mi
#pragma unroll
  for (int j = 0; j < 8; ++j)
    Pb[(size_t)(m0 + j + 8 * half) * 16 + mi] = acc[j];
}

// ---------------------------------------------------------------------------
// Per-row coordinate-descent recurrence:
// a_r = (P[m,r] - sum_{s>r} u[m,s]*G[s,r] - sum_{s<r} a[m,s]*G[s,r] + EPS)
//       / (G[r,r] + EPS)
__global__ void __launch_bounds__(256)
cd_recurrence(const float* __restrict__ P, const float* __restrict__ Uold,
              const float* __restrict__ G, float* __restrict__ Unew, int Mo) {
  const int b = blockIdx.y;
  const int m = blockIdx.x * 256 + threadIdx.x;
  __shared__ float g[256];
  g[threadIdx.x] = G[b * 256 + threadIdx.x];
  __syncthreads();

  const float* p   = P    + ((size_t)b * Mo + m) * 16;
  const float* u   = Uold + ((size_t)b * Mo + m) * 16;
  float*       out = Unew + ((size_t)b * Mo + m) * 16;

  float pv[16], uv[16], av[16];
#pragma unroll
  for (int i = 0; i < 16; ++i) { pv[i] = p[i]; uv[i] = u[i]; }

#pragma unroll
  for (int r = 0; r < 16; ++r) {
    float num = pv[r] + EPSF;
#pragma unroll
    for (int s = r + 1; s < 16; ++s) num -= uv[s] * g[s * 16 + r];
#pragma unroll
    for (int s = 0; s < r; ++s)      num -= av[s] * g[s * 16 + r];
    av[r] = num / (g[r * 16 + r] + EPSF);
  }
#pragma unroll
  for (int i = 0; i < 16; ++i) out[i] = av[i];
}

// ---------------------------------------------------------------------------
extern "C" void kernel_launch(void* const* d_in, const int* in_sizes, int n_in,
                              void* d_out, int out_size, void* d_ws, size_t ws_size,
                              hipStream_t stream) {
  (void)in_sizes; (void)n_in; (void)out_size; (void)ws_size;

  const float* x = (const float*)d_in[0];   // [B, M, N]
  const float* u = (const float*)d_in[1];   // [B, M, R]
  const float* v = (const float*)d_in[2];   // [B, N, R]

  float* u_new = (float*)d_out;                       // [B, M, R]
  float* v_new = u_new + (size_t)Bc * Mc * Rr;        // [B, N, R]

  float* ws = (float*)d_ws;
  float* gV = ws;                                     // [B, 16, 16]
  float* gU = gV + Bc * 256;                          // [B, 16, 16]
  float* P  = gU + Bc * 256;                          // [B, M, 16]
  float* P2 = P  + (size_t)Bc * Mc * Rr;              // [B, N, 16]

  // u-update: P = X @ V, G = V^T V, recurrence -> u_new
  gram16<<<dim3(Bc), 256, 0, stream>>>(v, gV, Nc);
  gemm_x_w<false><<<dim3(Mc / ROWS_PER_WG, Bc), 256, 0, stream>>>(x, v, P, Mc, Nc);
  cd_recurrence<<<dim3(Mc / 256, Bc), 256, 0, stream>>>(P, u, gV, u_new, Mc);

  // v-update: P2 = X^T @ u_new, G = u_new^T u_new, recurrence -> v_new
  gram16<<<dim3(Bc), 256, 0, stream>>>(u_new, gU, Mc);
  gemm_x_w<true><<<dim3(Nc / ROWS_PER_WG, Bc), 256, 0, stream>>>(x, u_new, P2, Nc, Mc);
  cd_recurrence<<<dim3(Nc / 256, Bc), 256, 0, stream>>>(P2, v, gU, v_new, Nc);
}